// MLKD_Loss_13546326851608
// MI455X (gfx1250) — compile-verified
//
#include <hip/hip_runtime.h>
#include <cstdint>

#define L_ 12
#define B_ 16
#define NH_ 12
#define S_ 128
#define H_ 768
#define K_ 16
#define NC_ 2
#define ALPHA_ 0.1f
#define BETA_ 0.1f

typedef __attribute__((ext_vector_type(2))) float v2f;
typedef __attribute__((ext_vector_type(8))) float v8f;

__device__ __forceinline__ int iclamp(int v, int lo, int hi) {
    return v < lo ? lo : (v > hi ? hi : v);
}

// ---------------------------------------------------------------------------
// ws[0] = hidden squared-error accumulator, ws[1] = attention squared-error
// ---------------------------------------------------------------------------
__global__ void k_zero(float* ws) {
    if (threadIdx.x < 2) ws[threadIdx.x] = 0.0f;
}

// ---------------------------------------------------------------------------
// Hidden loss: one block per (l,b,k). Gather teacher/student span rows
// (<=8 x 768 f32 each) into LDS via CDNA5 async-to-LDS, then column-sum,
// scale by 1/len, squared diff over H, block-reduce, atomicAdd.
// ---------------------------------------------------------------------------
__global__ __launch_bounds__(256) void k_hidden(const float* __restrict__ th,
                                                const float* __restrict__ sh,
                                                const int* __restrict__ ta,
                                                const int* __restrict__ sa,
                                                const int* __restrict__ cslen,
                                                float* __restrict__ ws) {
    __shared__ float lds[16 * H_];   // 8 teacher rows + 8 student rows (48 KB)
    __shared__ float red[8];

    int bid = blockIdx.x;
    int k = bid % K_;
    int b = (bid / K_) % B_;
    int l = bid / (K_ * B_);

    int clen = iclamp(cslen[b], 0, K_);
    if (k >= clen) return;           // uniform: masked span contributes 0

    int t_s = ta[(b * K_ + k) * 2 + 0];
    int t_e = ta[(b * K_ + k) * 2 + 1];
    int s_s = sa[(b * K_ + k) * 2 + 0];
    int s_e = sa[(b * K_ + k) * 2 + 1];
    int lt = iclamp(t_e - t_s, 1, 8);
    int ls = iclamp(s_e - s_s, 1, 8);
    t_s = iclamp(t_s, 0, S_ - lt);
    s_s = iclamp(s_s, 0, S_ - ls);

    const float* tb = th + ((size_t)((l * B_ + b) * S_ + t_s)) * H_;
    const float* sb = sh + ((size_t)((l * B_ + b) * S_ + s_s)) * H_;

    unsigned ldsBase = (unsigned)(size_t)(&lds[0]);  // low 32 bits of flat LDS addr = LDS offset
    const int CH = (H_ * 4) / 16;                    // 192 b128 chunks per row
    int tot = (lt + ls) * CH;
    for (int c = threadIdx.x; c < tot; c += 256) {
        int r = c / CH;
        int o = (c % CH) * 16;                        // byte offset within row
        const char* g;
        unsigned dsa;
        if (r < lt) {
            g = (const char*)tb + (size_t)r * (H_ * 4) + o;
            dsa = ldsBase + (unsigned)(r * (H_ * 4) + o);
        } else {
            int r2 = r - lt;
            g = (const char*)sb + (size_t)r2 * (H_ * 4) + o;
            dsa = ldsBase + (unsigned)((8 + r2) * (H_ * 4) + o);
        }
        // CDNA5 async DMA: global -> LDS, tracked by ASYNCcnt
        asm volatile("global_load_async_to_lds_b128 %0, %1, off"
                     :: "v"(dsa), "v"((uint64_t)(size_t)g)
                     : "memory");
    }
    asm volatile("s_wait_asynccnt 0" ::: "memory");
    __syncthreads();

    float rlt = 1.0f / (float)lt;
    float rls = 1.0f / (float)ls;
    float sq = 0.0f;
#pragma unroll
    for (int m = 0; m < 3; ++m) {
        int col = threadIdx.x + 256 * m;
        float at = 0.0f, as = 0.0f;
        for (int j = 0; j < lt; ++j) at += lds[j * H_ + col];
        for (int j = 0; j < ls; ++j) as += lds[(8 + j) * H_ + col];
        float d = at * rlt - as * rls;
        sq += d * d;
    }

    // wave32 reduce then cross-wave reduce
    for (int off = 16; off > 0; off >>= 1) sq += __shfl_down(sq, off, 32);
    if ((threadIdx.x & 31) == 0) red[threadIdx.x >> 5] = sq;
    __syncthreads();
    if (threadIdx.x < 8) {
        float v = red[threadIdx.x];
        for (int off = 4; off > 0; off >>= 1) v += __shfl_down(v, off, 8);
        if (threadIdx.x == 0) atomicAdd(&ws[0], v);
    }
}

// ---------------------------------------------------------------------------
// Attention loss: one block per (l,b,nh), 256 threads = 8 waves.
// Gather masked span rows into LDS G[128x128] (16 spans x 8 padded rows),
// then diff = Wt*Gt - Ws*Gs computed with chained V_WMMA_F32_16X16X4_F32;
// each wave owns one 16-column tile. Invalid spans are zero-padded so the
// mask is implicit; final masked Frobenius norm reduced per wave.
// ---------------------------------------------------------------------------
__global__ __launch_bounds__(256) void k_att(const float* __restrict__ tA,
                                             const float* __restrict__ sA,
                                             const int* __restrict__ ta,
                                             const int* __restrict__ sa,
                                             const int* __restrict__ cslen,
                                             float* __restrict__ ws) {
    __shared__ float G[S_ * S_];  // 64 KB gather buffer

    int bid = blockIdx.x;
    int nh = bid % NH_;
    int b = (bid / NH_) % B_;
    int l = bid / (NH_ * B_);

    int clen = iclamp(cslen[b], 0, K_);
    if (clen == 0) return;        // uniform: whole batch row masked

    int tid = threadIdx.x;
    int lane = tid & 31;
    int wave = tid >> 5;          // tile id: columns [wave*16, wave*16+16)
    int lk = lane & 15;

    // per-span reciprocal lengths, held in lane lk (lanes k and k+16 hold span k)
    int tt0 = ta[(b * K_ + lk) * 2 + 0], tt1 = ta[(b * K_ + lk) * 2 + 1];
    int ss0 = sa[(b * K_ + lk) * 2 + 0], ss1 = sa[(b * K_ + lk) * 2 + 1];
    float rt = 1.0f / (float)iclamp(tt1 - tt0, 1, S_);
    float rs = 1.0f / (float)iclamp(ss1 - ss0, 1, S_);

    const float* baseT = tA + ((size_t)((l * B_ + b) * NH_ + nh)) * S_ * S_;
    const float* baseS = sA + ((size_t)((l * B_ + b) * NH_ + nh)) * S_ * S_;

    auto gather = [&](const float* base, const int* al) {
        // 256 threads cover 2 rows x 128 cols per iteration; 64 iterations
        for (int it = 0; it < 64; ++it) {
            int r = it * 2 + (tid >> 7);   // 0..127 : span k = r>>3, pad row j = r&7
            int c = tid & 127;
            int k = r >> 3;
            int j = r & 7;
            float v = 0.0f;
            if (k < clen) {
                int st = al[(b * K_ + k) * 2 + 0];
                int en = al[(b * K_ + k) * 2 + 1];
                int len = en - st;
                if (j < len) {
                    int row = iclamp(st + j, 0, S_ - 1);
                    float a = base[(size_t)row * S_ + c];
                    v = (a <= -100.0f) ? 0.0f : a;   // zero masked (-inf style) entries
                }
            }
            G[r * S_ + c] = v;
        }
    };

    int n = lane & 15;       // column within the wave's 16-col tile
    int ko = lane >> 4;      // 0/1 : which K row of the B fragment pair
    v8f acc = {};            // D/C accumulator: 16(spans) x 16(cols) tile

    // ---- teacher pass: acc += Wt * Gt -------------------------------------
    gather(baseT, ta);
    __syncthreads();
    for (int q = 0; q < 32; ++q) {
        int m0 = q >> 1;                           // only nonzero weight row
        float w = __shfl(rt, m0, 32);              // broadcast 1/len_t(k=m0)
        float av = (lk == m0) ? w : 0.0f;
        v2f a = {av, av};                          // A[16x4] fragment (block-diag W slice)
        v2f bf = {G[(4 * q + ko) * S_ + wave * 16 + n],
                  G[(4 * q + 2 + ko) * S_ + wave * 16 + n]};
        acc = __builtin_amdgcn_wmma_f32_16x16x4_f32(false, a, false, bf,
                                                    (short)0, acc, false, false);
    }
    __syncthreads();

    // ---- student pass: acc += (-Ws) * Gs  =>  acc = diff ------------------
    gather(baseS, sa);
    __syncthreads();
    for (int q = 0; q < 32; ++q) {
        int m0 = q >> 1;
        float w = -__shfl(rs, m0, 32);             // negate 1/len_s(k=m0)
        float av = (lk == m0) ? w : 0.0f;
        v2f a = {av, av};
        v2f bf = {G[(4 * q + ko) * S_ + wave * 16 + n],
                  G[(4 * q + 2 + ko) * S_ + wave * 16 + n]};
        acc = __builtin_amdgcn_wmma_f32_16x16x4_f32(false, a, false, bf,
                                                    (short)0, acc, false, false);
    }

    // masked sum of squares: VGPR i of lane holds diff[k = i + 8*ko, col]
    float sq = 0.0f;
#pragma unroll
    for (int i = 0; i < 8; ++i) {
        int kk = i + 8 * ko;
        float d = acc[i];
        if (kk < clen) sq += d * d;
    }
    for (int off = 16; off > 0; off >>= 1) sq += __shfl_down(sq, off, 32);
    if (lane == 0) atomicAdd(&ws[1], sq);
}

// ---------------------------------------------------------------------------
// Finalize: n_spans, loss normalization, 16x2 cross-entropy.
// ---------------------------------------------------------------------------
__global__ void k_fin(const float* __restrict__ vl, const int* __restrict__ tgt,
                      const int* __restrict__ cslen, const float* __restrict__ ws,
                      float* __restrict__ out) {
    if (threadIdx.x != 0) return;
    int ns = 0;
    for (int b = 0; b < B_; ++b) ns += iclamp(cslen[b], 0, K_);
    float nsf = (float)ns;

    float hid = 0.0f, att = 0.0f;
    if (ns > 0) {
        hid = ALPHA_ * ws[0] / fmaxf(nsf * (float)(L_ * H_), 1.0f);
        att = BETA_ * ws[1] / fmaxf(nsf * (float)(L_ * NH_ * S_), 1.0f);
    }

    float pl = 0.0f;
    for (int b = 0; b < B_; ++b) {
        float x0 = vl[b * NC_ + 0];
        float x1 = vl[b * NC_ + 1];
        float mx = fmaxf(x0, x1);
        float lse = mx + __logf(__expf(x0 - mx) + __expf(x1 - mx));
        int t = iclamp(tgt[b], 0, NC_ - 1);
        float xt = vl[b * NC_ + t];
        pl += (lse - xt);
    }
    pl /= (float)B_;

    out[0] = hid;
    out[1] = att;
    out[2] = pl;
}

// ---------------------------------------------------------------------------
extern "C" void kernel_launch(void* const* d_in, const int* in_sizes, int n_in,
                              void* d_out, int out_size, void* d_ws, size_t ws_size,
                              hipStream_t stream) {
    const float* voted = (const float*)d_in[0];
    const int*   target = (const int*)d_in[1];
    const float* t_hid = (const float*)d_in[2];
    const float* t_att = (const float*)d_in[3];
    const float* s_hid = (const float*)d_in[4];
    const float* s_att = (const float*)d_in[5];
    const int*   t_al = (const int*)d_in[6];
    const int*   s_al = (const int*)d_in[7];
    const int*   cl = (const int*)d_in[8];
    float* ws = (float*)d_ws;
    float* out = (float*)d_out;
    (void)in_sizes; (void)n_in; (void)out_size; (void)ws_size;

    hipLaunchKernelGGL(k_zero, dim3(1), dim3(32), 0, stream, ws);
    hipLaunchKernelGGL(k_hidden, dim3(L_ * B_ * K_), dim3(256), 0, stream,
                       t_hid, s_hid, t_al, s_al, cl, ws);
    hipLaunchKernelGGL(k_att, dim3(L_ * B_ * NH_), dim3(256), 0, stream,
                       t_att, s_att, t_al, s_al, cl, ws);
    hipLaunchKernelGGL(k_fin, dim3(1), dim3(32), 0, stream,
                       voted, target, cl, ws, out);
}